// GCN_6365141533332
// MI455X (gfx1250) — compile-verified
//
#include <hip/hip_runtime.h>
#include <hip/hip_bf16.h>

// ---------------------------------------------------------------------------
// 2-layer GCN forward for gfx1250 (MI455X).
//  - Dense GEMMs: V_WMMA_F32_16X16X4_F32 (exact fp32 matrix op, CDNA5).
//  - Edge norm precomputed once (removes repeated random dis[] gathers).
//  - Scatter: half-wave per edge, float4 gather (b128) + f32 global atomics.
// ---------------------------------------------------------------------------

typedef __attribute__((ext_vector_type(2))) float v2f;
typedef __attribute__((ext_vector_type(8))) float v8f;

#define IN_C  128
#define HID_C 64
#define OUT_C 64

// ---------------------------- degree pipeline ------------------------------

__global__ void k_deg_init(float* __restrict__ deg, int n) {
    int i = blockIdx.x * blockDim.x + threadIdx.x;
    if (i < n) deg[i] = 1.0f;  // self-loop weight
}

__global__ void k_deg_accum(const long long* __restrict__ col,
                            const float* __restrict__ w,
                            float* __restrict__ deg, int nE) {
    int e = blockIdx.x * blockDim.x + threadIdx.x;
    if (e < nE) atomicAdd(&deg[(int)col[e]], w[e]);
}

__global__ void k_deg_finish(float* __restrict__ dis, int n) {
    int i = blockIdx.x * blockDim.x + threadIdx.x;
    if (i < n) {
        float d = dis[i];
        dis[i] = (d > 0.0f) ? rsqrtf(d) : 0.0f;
    }
}

// norm[e] = dis[row]*w[e]*dis[col]  (one coalesced pass; reused by both layers)
__global__ void k_edge_norm(const long long* __restrict__ ei,
                            const float* __restrict__ w,
                            const float* __restrict__ dis,
                            float* __restrict__ norm, int nE) {
    int e = blockIdx.x * blockDim.x + threadIdx.x;
    if (e < nE) {
        int r = (int)ei[e];
        int c = (int)ei[(size_t)nE + e];
        norm[e] = dis[r] * w[e] * dis[c];
    }
}

// ------------------------------ WMMA GEMM ----------------------------------
// C[M, 64] = A[M, K] @ W[K, 64]   (row-major, fp32, exact)
// One block = 128 threads = 4 waves; wave w computes the 16x16 tile at
// (rows blockIdx.x*16, cols 16*w) via a K-loop of 16x16x4 f32 WMMAs.
//
// f32 WMMA VGPR layouts (ISA 7.12.2):
//   A 16x4 : lane l (l<16) rows M=l,  VGPR j holds K = k0 + 2*(l>>4) + j
//   B 4x16 : lane l cols N=l&15,      VGPR j holds K = k0 + 2*(l>>4) + j
//   C 16x16: VGPR r = row (m0 + r + 8*(l>>4)), col n0 + (l&15)
template <int K>
__global__ __launch_bounds__(128) void k_gemm_wmma(const float* __restrict__ A,
                                                   const float* __restrict__ W,
                                                   float* __restrict__ C, int M) {
    const int wave = threadIdx.x >> 5;
    const int lane = threadIdx.x & 31;
    const int lm = lane & 15;   // row within A-tile / col within B-tile
    const int lh = lane >> 4;   // K-half selector
    const int m0 = blockIdx.x * 16;
    const int n0 = wave * 16;

    int arow = m0 + lm;
    if (arow >= M) arow = M - 1;              // clamp (no effect when M%16==0)
    const float* __restrict__ ap = A + (size_t)arow * K;
    const float* __restrict__ wp = W + (size_t)(2 * lh) * 64 + n0 + lm;

    v8f acc = {};
#pragma unroll
    for (int k0 = 0; k0 < K; k0 += 4) {
        v2f a, b;
        a.x = ap[k0 + 2 * lh + 0];
        a.y = ap[k0 + 2 * lh + 1];
        b.x = wp[(size_t)(k0 + 0) * 64];
        b.y = wp[(size_t)(k0 + 1) * 64];
        acc = __builtin_amdgcn_wmma_f32_16x16x4_f32(
            /*neg_a=*/false, a, /*neg_b=*/false, b,
            /*c_mod=*/(short)0, acc, /*reuse_a=*/false, /*reuse_b=*/false);
    }

    const int rbase = m0 + 8 * lh;
#pragma unroll
    for (int r = 0; r < 8; ++r) {
        int row = rbase + r;
        if (row < M) C[(size_t)row * 64 + n0 + lm] = acc[r];
    }
}

// --------------------- aggregation init: self-msg + bias -------------------
// agg[i][c] = dis[i]^2 * t[i][c] + bias[c]   (self-loop message folded in)
__global__ void k_self_bias_init(const float* __restrict__ t,
                                 const float* __restrict__ dis,
                                 const float* __restrict__ bias,
                                 float* __restrict__ agg, int n) {
    int idx = blockIdx.x * blockDim.x + threadIdx.x;
    if (idx >= n * 64) return;
    int i = idx >> 6;
    int c = idx & 63;
    float d = dis[i];
    agg[idx] = d * d * t[idx] + bias[c];
}

// ------------------------------ edge scatter -------------------------------
// Half-wave (16 lanes) per edge; lane j moves channels 4j..4j+3 (b128 gather).
// agg[col] += h[row] * norm[e]  via f32 global atomics (L2 atomic units).
__global__ __launch_bounds__(256) void k_scatter(const long long* __restrict__ ei,
                                                 const float* __restrict__ norm,
                                                 const float* __restrict__ h,
                                                 float* __restrict__ agg, int nE) {
    int e = blockIdx.x * 16 + (threadIdx.x >> 4);
    if (e >= nE) return;
    int j = threadIdx.x & 15;

    // stay ahead of the sequential index/norm streams (global_prefetch_b8)
    __builtin_prefetch(&ei[e + 2048], 0, 1);
    __builtin_prefetch(&norm[e + 4096], 0, 1);

    int r = (int)ei[e];
    int c = (int)ei[(size_t)nE + e];
    float nrm = norm[e];

    float4 hv = ((const float4*)(h + (size_t)r * 64))[j];
    float* __restrict__ ac = agg + (size_t)c * 64 + 4 * j;
    atomicAdd(ac + 0, hv.x * nrm);
    atomicAdd(ac + 1, hv.y * nrm);
    atomicAdd(ac + 2, hv.z * nrm);
    atomicAdd(ac + 3, hv.w * nrm);
}

// --------------------------------- ReLU ------------------------------------
__global__ void k_relu(float* __restrict__ a, int count) {
    int i = blockIdx.x * blockDim.x + threadIdx.x;
    if (i < count) a[i] = fmaxf(a[i], 0.0f);
}

// ------------------------------- launcher ----------------------------------

extern "C" void kernel_launch(void* const* d_in, const int* in_sizes, int n_in,
                              void* d_out, int out_size, void* d_ws, size_t ws_size,
                              hipStream_t stream) {
    const float*     x  = (const float*)d_in[0];      // [N, 128]
    const long long* ei = (const long long*)d_in[1];  // [2, nE] int64
    const float*     ew = (const float*)d_in[2];      // [nE]
    const float*     W1 = (const float*)d_in[3];      // [128, 64]
    const float*     b1 = (const float*)d_in[4];      // [64]
    const float*     W2 = (const float*)d_in[5];      // [64, 64]
    const float*     b2 = (const float*)d_in[6];      // [64]
    float*           out = (float*)d_out;             // [N, 64]

    const int N  = in_sizes[0] / IN_C;
    const int nE = in_sizes[2];

    // Workspace layout: dis[N] | h[N*64] | agg[N*64] | norm[nE]
    char* ws = (char*)d_ws;
    size_t off = 0;
    float* dis  = (float*)(ws + off); off += ((size_t)N * 4 + 255) & ~(size_t)255;
    float* h    = (float*)(ws + off); off += (size_t)N * 64 * 4;
    float* agg  = (float*)(ws + off); off += (size_t)N * 64 * 4;
    float* norm = (float*)(ws + off); off += (size_t)nE * 4;
    (void)ws_size; (void)n_in; (void)out_size;

    const int T = 256;
    const int gN  = (N + T - 1) / T;
    const int gE  = (nE + T - 1) / T;
    const int gNC = (N * 64 + T - 1) / T;
    const int gES = (nE + 15) / 16;      // 16 edges per 256-thread block
    const int gM  = (N + 15) / 16;       // GEMM row tiles

    // ---- degree / normalization (shared by both layers) ----
    k_deg_init<<<gN, T, 0, stream>>>(dis, N);
    k_deg_accum<<<gE, T, 0, stream>>>(ei + nE, ew, dis, nE);  // col = ei[1]
    k_deg_finish<<<gN, T, 0, stream>>>(dis, N);
    k_edge_norm<<<gE, T, 0, stream>>>(ei, ew, dis, norm, nE);

    // ---- layer 1: h = x @ W1 ; agg = scatter(norm * h) + b1 ; relu ----
    k_gemm_wmma<IN_C><<<gM, 128, 0, stream>>>(x, W1, h, N);
    k_self_bias_init<<<gNC, T, 0, stream>>>(h, dis, b1, agg, N);
    k_scatter<<<gES, T, 0, stream>>>(ei, norm, h, agg, nE);
    k_relu<<<gNC, T, 0, stream>>>(agg, N * 64);

    // ---- layer 2: t = relu(agg) @ W2 ; out = scatter(norm * t) + b2 ----
    k_gemm_wmma<HID_C><<<gM, 128, 0, stream>>>(agg, W2, h, N);  // reuse h as t
    k_self_bias_init<<<gNC, T, 0, stream>>>(h, dis, b2, out, N);
    k_scatter<<<gES, T, 0, stream>>>(ei, norm, h, out, nE);
}